// LightGCN_66460323938526
// MI455X (gfx1250) — compile-verified
//
#include <hip/hip_runtime.h>

#define EMBED_DIM 64
#define NUM_LAYER 3

typedef __attribute__((ext_vector_type(2))) float v2f;
typedef __attribute__((ext_vector_type(8))) float v8f;

// ---------------------------------------------------------------------------
// init: x = concat(user_emb, item_emb); acc = x; y = 0
// ---------------------------------------------------------------------------
__global__ __launch_bounds__(256) void lgcn_init_kernel(
    const float* __restrict__ ue, const float* __restrict__ ie,
    float* __restrict__ x, float* __restrict__ acc, float* __restrict__ y,
    long long n_user_elems, long long n_total) {
  long long i = (long long)blockIdx.x * blockDim.x + threadIdx.x;
  if (i >= n_total) return;
  float v = (i < n_user_elems) ? ue[i] : ie[i - n_user_elems];
  x[i]   = v;
  acc[i] = v;
  y[i]   = 0.0f;
}

// ---------------------------------------------------------------------------
// SpMM scatter: y[rows[e], :] += vals[e] * x[cols[e], :]
// 16 consecutive lanes per edge -> 256B coalesced row gather,
// global_atomic_add_f32 (no-return, STOREcnt path) scatter through L2.
// Entire x/y tables (38.4MB each) are L2-resident on MI455X (192MB L2).
// ---------------------------------------------------------------------------
__global__ __launch_bounds__(256) void lgcn_spmm_kernel(
    const float* __restrict__ x, float* __restrict__ y,
    const float* __restrict__ vals, const int* __restrict__ rows,
    const int* __restrict__ cols, int nnz) {
  long long t = (long long)blockIdx.x * blockDim.x + threadIdx.x;
  int e = (int)(t >> 4);
  if (e >= nnz) return;
  int d4 = (int)(t & 15);

  int   r = rows[e];
  int   c = cols[e];
  float v = vals[e];

  const float4 xv = ((const float4*)(x + (size_t)c * EMBED_DIM))[d4];
  float* yp = y + (size_t)r * EMBED_DIM + (size_t)d4 * 4;
  unsafeAtomicAdd(yp + 0, v * xv.x);
  unsafeAtomicAdd(yp + 1, v * xv.y);
  unsafeAtomicAdd(yp + 2, v * xv.z);
  unsafeAtomicAdd(yp + 3, v * xv.w);
}

// ---------------------------------------------------------------------------
// update: acc += y ; zero the buffer that will be next layer's destination
// ---------------------------------------------------------------------------
__global__ __launch_bounds__(256) void lgcn_update_kernel(
    float* __restrict__ acc, const float* __restrict__ nxt,
    float* __restrict__ tozero, long long n_total) {
  long long i = (long long)blockIdx.x * blockDim.x + threadIdx.x;
  if (i >= n_total) return;
  acc[i] += nxt[i];
  tozero[i] = 0.0f;
}

// ---------------------------------------------------------------------------
// Scorer: one wave (32 lanes) per 16 batch elements.
// D[m,n] = sum_k U[m,k] * I[n,k] via 16 chained V_WMMA_F32_16X16X4_F32
// (K accumulated 4 at a time over 64 dims, C=D accumulation).
// scores = diag(D) * (1/16)   [emb = acc/4 for both sides]
//
// A 16x4 f32 layout: lane l holds M = l%16, K = 2*(l/16) + {0,1} in 2 VGPRs.
// B  4x16 f32 layout: lane l holds N = l%16, K = 2*(l/16) + {0,1} — i.e. the
// same (lane -> row, K-pair) mapping, so both tiles load identically: a
// float2 at row[2*(kb/2) + 2*(l/16)].
// C/D 16x16 f32 layout: VGPR j, lane l -> (M = j + 8*(l/16), N = l%16).
// Diagonal: n<8 at lane n, c[n]; n>=8 at lane n+16, c[n-8].
// ---------------------------------------------------------------------------
__global__ __launch_bounds__(32) void lgcn_score_wmma_kernel(
    const float* __restrict__ acc, const int* __restrict__ user_ids,
    const int* __restrict__ item_ids, float* __restrict__ out, int num_user) {
  const int g     = blockIdx.x;        // group of 16 batch elements
  const int l     = threadIdx.x;       // lane 0..31 (wave32)
  const int m     = l & 15;            // tile row (M for A, N for B)
  const int khalf = l >> 4;            // which K-pair this lane holds

  const int b  = g * 16 + m;
  const int u  = user_ids[b];
  const int it = item_ids[b];

  const v2f* __restrict__ u2 = (const v2f*)(acc + (size_t)u * EMBED_DIM);
  const v2f* __restrict__ i2 = (const v2f*)(acc + (size_t)(num_user + it) * EMBED_DIM);

  v8f c = {};
#pragma unroll
  for (int kb = 0; kb < EMBED_DIM; kb += 4) {
    v2f av = u2[(kb >> 1) + khalf];    // U[m, kb + 2*khalf + {0,1}]
    v2f bv = i2[(kb >> 1) + khalf];    // I[n, kb + 2*khalf + {0,1}]
    // D = A x B + C  (A: 16x4, B: 4x16, C/D: 16x16 f32)
    c = __builtin_amdgcn_wmma_f32_16x16x4_f32(
        /*neg_a=*/false, av, /*neg_b=*/false, bv,
        /*c_mod=*/(short)0, c, /*reuse_a=*/false, /*reuse_b=*/false);
  }

  // Extract diagonal and apply (1/4)*(1/4) layer-mean normalization.
  if (l < 8) {
    out[g * 16 + l] = c[l] * 0.0625f;
  } else if (l >= 24) {
    out[g * 16 + (l - 16)] = c[l - 24] * 0.0625f;
  }
}

// ---------------------------------------------------------------------------
// Host-side orchestration (graph-capture safe: only async launches on stream)
// ---------------------------------------------------------------------------
extern "C" void kernel_launch(void* const* d_in, const int* in_sizes, int n_in,
                              void* d_out, int out_size, void* d_ws, size_t ws_size,
                              hipStream_t stream) {
  const float* user_emb = (const float*)d_in[0];
  const float* item_emb = (const float*)d_in[1];
  const float* adj_vals = (const float*)d_in[2];
  const int*   adj_rows = (const int*)d_in[3];
  const int*   adj_cols = (const int*)d_in[4];
  const int*   user_ids = (const int*)d_in[5];
  const int*   item_ids = (const int*)d_in[6];
  float*       out      = (float*)d_out;

  const long long n_user_elems = (long long)in_sizes[0];          // NUM_USER * 64
  const long long n_item_elems = (long long)in_sizes[1];          // NUM_ITEM * 64
  const long long n_total      = n_user_elems + n_item_elems;     // N * 64
  const int       nnz          = in_sizes[2];
  const int       batch        = in_sizes[5];
  const int       num_user     = (int)(n_user_elems / EMBED_DIM);

  // Workspace: 3 x [N*64] f32 (b0, b1, acc) = ~115 MB
  float* b0  = (float*)d_ws;
  float* b1  = b0 + n_total;
  float* acc = b1 + n_total;

  const int ethreads = 256;
  const int eblocks  = (int)((n_total + ethreads - 1) / ethreads);

  lgcn_init_kernel<<<eblocks, ethreads, 0, stream>>>(
      user_emb, item_emb, b0, acc, b1, n_user_elems, n_total);

  float* cur = b0;
  float* nxt = b1;
  const long long sp_threads = (long long)nnz * 16;
  const int sp_blocks = (int)((sp_threads + 255) / 256);

  for (int layer = 0; layer < NUM_LAYER; ++layer) {
    lgcn_spmm_kernel<<<sp_blocks, 256, 0, stream>>>(
        cur, nxt, adj_vals, adj_rows, adj_cols, nnz);
    // acc += nxt; zero `cur` so it is a clean destination for the next layer
    lgcn_update_kernel<<<eblocks, ethreads, 0, stream>>>(acc, nxt, cur, n_total);
    float* tmp = cur; cur = nxt; nxt = tmp;
  }

  // One wave per 16 batch elements: 4096/16 = 256 blocks of 32 threads.
  lgcn_score_wmma_kernel<<<batch / 16, 32, 0, stream>>>(
      acc, user_ids, item_ids, out, num_user);
}